// YoshidaIntegrator_2370821948213
// MI455X (gfx1250) — compile-verified
//
#include <hip/hip_runtime.h>

typedef float v2f __attribute__((ext_vector_type(2)));
typedef float v8f __attribute__((ext_vector_type(8)));

#define BD 64
#define RR 16

__device__ __forceinline__ v8f wmma4(v2f a, v2f b, v8f c) {
  // V_WMMA_F32_16X16X4_F32: D = A(16x4) x B(4x16) + C(16x16), f32
  return __builtin_amdgcn_wmma_f32_16x16x4_f32(false, a, false, b, (short)0, c,
                                               false, false);
}

__device__ __forceinline__ void lds_fence() {
  asm volatile("s_wait_dscnt 0x0" ::: "memory");
}

__global__ __launch_bounds__(256) void yoshida4_kernel(
    const float* __restrict__ X, const float* __restrict__ V,
    const float* __restrict__ F, const float* __restrict__ U,
    const float* __restrict__ W, float* __restrict__ OX,
    float* __restrict__ OV, int nrows) {
  // Yoshida-4 coefficients (double, folded into f32 with dt)
  constexpr double cbrt2 = 1.2599210498948731647672106072782;
  constexpr double w1 = 1.0 / (2.0 - cbrt2);
  constexpr double w0 = -cbrt2 / (2.0 - cbrt2);
  constexpr double dt = 0.01;
  const float C1 = (float)(w1 * 0.5 * dt);          // = C4
  const float C2 = (float)((w0 + w1) * 0.5 * dt);   // = C3
  const float D1 = (float)(w1 * dt);                // = D3
  const float D2 = (float)(w0 * dt);

  const int lane = threadIdx.x & 31;
  const int wave = threadIdx.x >> 5;
  const int half = lane >> 4;   // 0: lanes 0-15, 1: lanes 16-31
  const int l16 = lane & 15;

  // Per-wave LDS staging for the C/D-layout -> A-layout transposes.
  constexpr int VSTRIDE = 68;  // 64 + 4 pad -> conflict-free both directions
  constexpr int HSTRIDE = 20;  // 16 + 4 pad
  constexpr int WAVE_LDS = 16 * VSTRIDE + 16 * HSTRIDE;  // 1408 floats
  __shared__ float lds[8 * WAVE_LDS];
  float* ldsV = &lds[wave * WAVE_LDS];
  float* ldsH = ldsV + 16 * VSTRIDE;

  const int row0 = ((int)blockIdx.x * 8 + wave) * 16;
  if (row0 >= nrows) return;

  // ---- Preload U (64x16) as B-operands for 16 k-steps of K=64 ----
  // B-layout 4x16: vgpr0 lanes0-15 K=0 / lanes16-31 K=2; vgpr1 K=1 / K=3
  v2f uB[16];
#pragma unroll
  for (int k = 0; k < 16; ++k) {
    const int r0 = 4 * k + 2 * half;
    uB[k].x = U[r0 * RR + l16];
    uB[k].y = U[(r0 + 1) * RR + l16];
  }
  // ---- Preload W (16x64): 4 output d-tiles x 4 k-steps of K=16 ----
  v2f wB[4][4];
#pragma unroll
  for (int n = 0; n < 4; ++n)
#pragma unroll
    for (int k = 0; k < 4; ++k) {
      const int r0 = 4 * k + 2 * half;
      wB[n][k].x = W[r0 * BD + 16 * n + l16];
      wB[n][k].y = W[(r0 + 1) * BD + 16 * n + l16];
    }

  // ---- Load state tiles in WMMA C/D layout ----
  // element (n,i): row b = row0 + i + 8*half, col d = 16n + l16
  v8f xt[4], vt[4], ft[4];
#pragma unroll
  for (int n = 0; n < 4; ++n)
#pragma unroll
    for (int i = 0; i < 8; ++i) {
      const size_t idx = (size_t)(row0 + i + 8 * half) * BD + 16 * n + l16;
      xt[n][i] = X[idx];
      vt[n][i] = V[idx];
      ft[n][i] = F[idx];
    }

  // One kick: v += ddt * (-(v@U)^2 @ W + force), evaluated at current v.
  auto kick = [&](float ddt) {
    // v tile (C/D layout regs) -> LDS row-major [16 x 64]
#pragma unroll
    for (int n = 0; n < 4; ++n)
#pragma unroll
      for (int i = 0; i < 8; ++i)
        ldsV[(i + 8 * half) * VSTRIDE + 16 * n + l16] = vt[n][i];
    lds_fence();

    // h = v @ U : 16 chained wmma (K=64)
    v8f h = {};
#pragma unroll
    for (int k = 0; k < 16; ++k) {
      const v2f a = *(const v2f*)&ldsV[l16 * VSTRIDE + 4 * k + 2 * half];
      h = wmma4(a, uB[k], h);
    }

    // h2s = -(ddt) * h*h  -> LDS transpose staging [16 x 16]
#pragma unroll
    for (int i = 0; i < 8; ++i) {
      const float t = h[i] * (-ddt);
      ldsH[(i + 8 * half) * HSTRIDE + l16] = t * h[i];
    }
    lds_fence();

    v2f aH[4];
#pragma unroll
    for (int k = 0; k < 4; ++k)
      aH[k] = *(const v2f*)&ldsH[l16 * HSTRIDE + 4 * k + 2 * half];

    // v = v + h2s @ W + ddt * force  (accumulate straight into v tiles)
#pragma unroll
    for (int n = 0; n < 4; ++n) {
      v8f c = vt[n];
#pragma unroll
      for (int k = 0; k < 4; ++k) c = wmma4(aH[k], wB[n][k], c);
#pragma unroll
      for (int i = 0; i < 8; ++i) c[i] += ddt * ft[n][i];
      vt[n] = c;
    }
  };

  auto drift = [&](float cdt) {
#pragma unroll
    for (int n = 0; n < 4; ++n)
#pragma unroll
      for (int i = 0; i < 8; ++i) xt[n][i] += cdt * vt[n][i];
  };

  drift(C1);
  kick(D1);
  drift(C2);
  kick(D2);
  drift(C2);  // C3 == C2
  kick(D1);   // D3 == D1
  drift(C1);  // C4 == C1

  // ---- Store x_final, v3 ----
#pragma unroll
  for (int n = 0; n < 4; ++n)
#pragma unroll
    for (int i = 0; i < 8; ++i) {
      const size_t idx = (size_t)(row0 + i + 8 * half) * BD + 16 * n + l16;
      OX[idx] = xt[n][i];
      OV[idx] = vt[n][i];
    }
}

extern "C" void kernel_launch(void* const* d_in, const int* in_sizes, int n_in,
                              void* d_out, int out_size, void* d_ws,
                              size_t ws_size, hipStream_t stream) {
  const float* X = (const float*)d_in[0];
  const float* V = (const float*)d_in[1];
  const float* F = (const float*)d_in[2];
  const float* U = (const float*)d_in[3];
  const float* W = (const float*)d_in[4];
  const int nrows = in_sizes[0] / BD;  // B = 1048576
  float* OX = (float*)d_out;
  float* OV = OX + (size_t)nrows * BD;

  const int rows_per_block = 8 * 16;  // 8 wave32s, 16-row tile each
  const int grid = (nrows + rows_per_block - 1) / rows_per_block;
  yoshida4_kernel<<<grid, 256, 0, stream>>>(X, V, F, U, W, OX, OV, nrows);
}